// FinalNMSLoss_7344394076342
// MI455X (gfx1250) — compile-verified
//
#include <hip/hip_runtime.h>

#ifndef __has_builtin
#define __has_builtin(x) 0
#endif

#if __has_builtin(__builtin_amdgcn_tensor_load_to_lds) && __has_builtin(__builtin_amdgcn_s_wait_tensorcnt)
#define HAVE_TDM 1
#else
#define HAVE_TDM 0
#endif

#if __has_builtin(__builtin_amdgcn_wmma_f32_16x16x4_f32)
#define HAVE_WMMA4 1
#else
#define HAVE_WMMA4 0
#endif

// Problem constants (match reference)
constexpr int Bc = 8;
constexpr int Nc = 1024;
constexpr int Mc = 20000;
constexpr int Cc = 80;
constexpr int Gc = 64;
constexpr float THR  = 0.5f;   // NMS_THR
constexpr float EPSF = 1e-6f;

constexpr int TPB  = 256;       // threads per block (8 wave32s)
constexpr int NWV  = TPB / 32;  // waves per block
constexpr int EPT  = Nc / TPB;  // 4 elements per thread

typedef __attribute__((ext_vector_type(4))) unsigned int u32x4;
typedef __attribute__((ext_vector_type(8))) int          i32x8;
typedef __attribute__((ext_vector_type(4))) int          i32x4;
typedef __attribute__((ext_vector_type(2))) float        v2f;
typedef __attribute__((ext_vector_type(8))) float        v8f;

// legacy +1 IoU, matches pairwise_iou in the reference
__device__ __forceinline__ float iou1(float4 a, float4 b) {
    float lx = fmaxf(a.x, b.x), ly = fmaxf(a.y, b.y);
    float rx = fminf(a.z, b.z), ry = fminf(a.w, b.w);
    float w  = fmaxf(rx - lx + 1.0f, 0.0f);
    float h  = fmaxf(ry - ly + 1.0f, 0.0f);
    float ov = w * h;
    float a1 = (a.z - a.x + 1.0f) * (a.w - a.y + 1.0f);
    float a2 = (b.z - b.x + 1.0f) * (b.w - b.y + 1.0f);
    return ov / (a1 + a2 - ov);
}

__device__ __forceinline__ unsigned long long shfl_xor_u64(unsigned long long v, int m) {
    unsigned lo = (unsigned)v, hi = (unsigned)(v >> 32);
    lo = __shfl_xor(lo, m, 32);
    hi = __shfl_xor(hi, m, 32);
    return ((unsigned long long)hi << 32) | lo;
}

__global__ __launch_bounds__(TPB) void nms_batch_kernel(
    const int* __restrict__ pos_inds, const int* __restrict__ pos_gt,
    const float4* __restrict__ gt_bboxes, const float* __restrict__ bbox_preds,
    const float* __restrict__ cls_scores, const int* __restrict__ gt_labels,
    float* __restrict__ ws) {

    __shared__ float4 s_boxes[Nc];                  // 16 KB (TDM destination)
    __shared__ float  s_scores[Nc];                 // 4 KB
    __shared__ int    s_gt[Nc];                     // 4 KB
    __shared__ float  s_gtiou[Gc * Gc];             // 16 KB
    __shared__ float4 s_gbox[Gc];                   // 1 KB
    __shared__ int    s_seen[Gc];                   // 256 B
    __shared__ unsigned long long redK[NWV];
    __shared__ float  redF[NWV];
    __shared__ int    redI[NWV];

    const int tid  = threadIdx.x;
    const int lane = tid & 31;
    const int wid  = tid >> 5;
    const int b    = blockIdx.x;

#if HAVE_TDM
    // --- Stage boxes [1024x4 f32 = 16KB] global->LDS via Tensor Data Mover ---
    if (tid < 32) {  // one wave issues (TDM ignores EXEC; issue once per wave)
        unsigned long long ga =
            (unsigned long long)(const void*)(bbox_preds + (size_t)b * Nc * 4);
        unsigned ldsb = (unsigned)(size_t)(&s_boxes[0]);
        u32x4 g0;
        g0[0] = 1u;                                     // count=1, user mode
        g0[1] = ldsb;                                   // lds_addr
        g0[2] = (unsigned)ga;                           // global_addr[31:0]
        g0[3] = (unsigned)((ga >> 32) & 0x01FFFFFFu) |  // global_addr[56:32]
                0x80000000u;                            // type=2 ("image")
        i32x8 g1;
        g1[0] = 0x00020000;            // data_size=4B, no mask/flags
        g1[1] = (int)(4096u << 16);    // tensor_dim0 = 4096 elements
        g1[2] = (int)(1u << 16);       // tensor_dim1 = 1
        g1[3] = (int)(4096u << 16);    // tile_dim0 = 4096
        g1[4] = 1;                     // tile_dim1 = 1
        g1[5] = 4096;                  // tensor_dim0_stride
        g1[6] = (int)(4096u << 16);    // tensor_dim1_stride[15:0]
        g1[7] = 0;
        i32x4 gz = {0, 0, 0, 0};
#if defined(__clang_major__) && (__clang_major__ >= 23)
        i32x8 gz8 = {0, 0, 0, 0, 0, 0, 0, 0};
        __builtin_amdgcn_tensor_load_to_lds(g0, g1, gz, gz, gz8, 0);
#else
        __builtin_amdgcn_tensor_load_to_lds(g0, g1, gz, gz, 0);
#endif
    }
#else
    for (int q = tid; q < Nc * 4; q += TPB)
        ((float*)s_boxes)[q] = bbox_preds[(size_t)b * Nc * 4 + q];
#endif

    // --- Init (overlaps with the DMA): gt boxes, seen[], score gather ---
    if (tid < Gc) {
        s_gbox[tid] = gt_bboxes[b * Gc + tid];
        s_seen[tid] = -1;
    }

    int   gidx[EPT];
    float scr[EPT];
    unsigned long long key[EPT];   // key==0 <=> element dead
    for (int r = 0; r < EPT; ++r) {
        int j  = tid + TPB * r;
        int pi = pos_inds[b * Nc + j];
        int gi = pos_gt[b * Nc + j];
        int lb = gt_labels[b * Gc + gi];
        float sc = cls_scores[((size_t)b * Mc + pi) * Cc + lb];
        gidx[r] = gi;
        scr[r]  = sc;
        s_scores[j] = sc;
        s_gt[j]     = gi;
        // monotone key: max score wins, ties -> lowest index (JAX argmax);
        // low half is never zero for a live element, so key==0 marks "dead".
        unsigned sb = __float_as_uint(sc);
        sb = (sb & 0x80000000u) ? ~sb : (sb | 0x80000000u);
        key[r] = ((unsigned long long)sb << 32) | (0xFFFFFFFFu - (unsigned)j);
    }

#if HAVE_TDM
    if (tid < 32) __builtin_amdgcn_s_wait_tensorcnt(0);
#endif
    __syncthreads();  // boxes + gbox visible

    float4 box[EPT];
    for (int r = 0; r < EPT; ++r) box[r] = s_boxes[tid + TPB * r];

    // gtiou [64x64] (detached in reference -> just values)
    for (int e = 0; e < (Gc * Gc) / TPB; ++e) {
        int idx = tid * ((Gc * Gc) / TPB) + e;
        s_gtiou[idx] = iou1(s_gbox[idx >> 6], s_gbox[idx & 63]);
    }
    __syncthreads();

    // --- Sequential NMS scan (pop in descending score order) ---
    // 2 barriers per pop step:
    //   barA: after key partials -> every thread self-reduces redK[] and
    //         decodes (i, g) locally (no broadcast barrier needed).
    //   barB: after sum partials -> thread0 runs scalar Phase C.
    // redK is protected without double-buffering: its next write happens only
    // after barB, once every thread has read it; redF/redI rewrites
    // happen even later (after barA + Phase B).
    float tpull = 0.0f, tpush = 0.0f;
    int   pcnt = 0, qcnt = 0;

    for (;;) {
        // Phase A: 64-bit max-reduce of keys == argmax over alive scores
        unsigned long long k = key[0];
        for (int r = 1; r < EPT; ++r)
            if (key[r] > k) k = key[r];
        for (int m = 16; m; m >>= 1) {
            unsigned long long o = shfl_xor_u64(k, m);
            if (o > k) k = o;
        }
        if (lane == 0) redK[wid] = k;
        __syncthreads();                               // barA
        unsigned long long kk = redK[0];
        for (int w = 1; w < NWV; ++w)
            if (redK[w] > kk) kk = redK[w];
        if (kk == 0ull) break;  // nothing alive -> remaining ref steps are no-ops
        const int i = (int)(0xFFFFFFFFu - (unsigned)kk);
        const int g = s_gt[i];
        const float4 bi = s_boxes[i];

        // Phase B: IoU row recomputed on the fly; push terms + counts
        float fsum = 0.0f;
        int   pk   = 0;  // (n_push << 16) | alive_count_after_pop
        for (int r = 0; r < EPT; ++r) {
            int  j = tid + TPB * r;
            bool a = key[r] != 0ull;
            if (j == i) { a = false; key[r] = 0ull; }
            float row = iou1(bi, box[r]);
            bool ov = a && (row > THR);
            bool pm = ov && (gidx[r] != g) && (row > s_gtiou[g * Gc + gidx[r]]);
            if (pm) fsum += -logf(1.0f - row) * scr[r];
            pk += (pm ? (1 << 16) : 0) + (a ? 1 : 0);
            if (ov) key[r] = 0ull;  // suppress after counting `remaining`
        }
        for (int m = 16; m; m >>= 1) {
            fsum += __shfl_xor(fsum, m, 32);
            pk   += __shfl_xor(pk, m, 32);
        }
        if (lane == 0) { redF[wid] = fsum; redI[wid] = pk; }
        __syncthreads();                               // barB

        // Phase C: scalar pop bookkeeping (thread 0) — exact reference order
        if (tid == 0) {
            float sum_terms = 0.0f;
            int   packed    = 0;
            for (int w = 0; w < NWV; ++w) { sum_terms += redF[w]; packed += redI[w]; }
            int n_push = packed >> 16;
            int acnt   = packed & 0xFFFF;
            int prev   = s_seen[g];
            bool has_seen = (prev >= 0);
            float pull = 0.0f;
            if (has_seen) {
                float si = iou1(s_boxes[prev], bi);
                pull = -logf(1.0f - THR + fmaxf(si, EPSF)) * s_scores[i];
            } else {
                s_seen[g] = i;
            }
            pcnt += has_seen ? 1 : 0;  // unconditional per pop (before `remaining` gate)
            if (acnt > 0) {            // remaining: alive after pop, before overlap removal
                if (has_seen) tpull += pull;
                if (n_push > 0) { tpush += sum_terms / (float)n_push; qcnt += n_push; }
            }
        }
        // no trailing barrier: next barA orders everything (see note above)
    }

    if (tid == 0) {
        ws[b]      = tpush / ((float)qcnt + EPSF);  // push
        ws[Bc + b] = tpull / ((float)pcnt + EPSF);  // pull
    }
}

// Batch mean on the matrix unit: A row0 = push[0..7], row1 = pull[0..7],
// B = ones -> D[m,0] = sum_k A[m,k]. Two chained f32 WMMAs cover K=0..3,4..7.
__global__ __launch_bounds__(32) void finalize_kernel(const float* __restrict__ ws,
                                                      float* __restrict__ out) {
#if HAVE_WMMA4
    int l  = threadIdx.x;
    int m  = l & 15;
    int kb = (l >> 4) << 1;              // VGPR pair holds K = kb, kb+1
    int   ro   = (m < 2) ? m * Bc : 0;   // row 0 -> push, row 1 -> pull
    float mask = (m < 2) ? 1.0f : 0.0f;  // rows 2..15 are zero
    v2f a0, a1, bb;
    a0.x = ws[ro + kb] * mask;
    a0.y = ws[ro + kb + 1] * mask;
    a1.x = ws[ro + 4 + kb] * mask;
    a1.y = ws[ro + 4 + kb + 1] * mask;
    bb.x = 1.0f; bb.y = 1.0f;            // B = all ones (layout-independent)
    v8f c = {};
    c = __builtin_amdgcn_wmma_f32_16x16x4_f32(false, a0, false, bb, (short)0, c, false, false);
    c = __builtin_amdgcn_wmma_f32_16x16x4_f32(false, a1, false, bb, (short)0, c, false, false);
    if (l == 0) {
        out[0] = c[0] * (1.0f / (float)Bc);  // mean(push) * PUSH_W
        out[1] = c[1] * (1.0f / (float)Bc);  // mean(pull) * PULL_W
    }
#else
    if (threadIdx.x == 0) {
        float ps = 0.0f, pl = 0.0f;
        for (int k2 = 0; k2 < Bc; ++k2) { ps += ws[k2]; pl += ws[Bc + k2]; }
        out[0] = ps * (1.0f / (float)Bc);
        out[1] = pl * (1.0f / (float)Bc);
    }
#endif
}

extern "C" void kernel_launch(void* const* d_in, const int* in_sizes, int n_in,
                              void* d_out, int out_size, void* d_ws, size_t ws_size,
                              hipStream_t stream) {
    (void)in_sizes; (void)n_in; (void)out_size; (void)ws_size;
    const int*    pos_inds  = (const int*)d_in[0];
    const int*    pos_gt    = (const int*)d_in[1];
    const float4* gt_bboxes = (const float4*)d_in[2];
    const float*  preds     = (const float*)d_in[3];
    const float*  cls       = (const float*)d_in[4];
    const int*    glab      = (const int*)d_in[5];
    float*        ws        = (float*)d_ws;
    float*        out       = (float*)d_out;

    nms_batch_kernel<<<Bc, TPB, 0, stream>>>(pos_inds, pos_gt, gt_bboxes, preds, cls, glab, ws);
    finalize_kernel<<<1, 32, 0, stream>>>(ws, out);
}